// AMS_new_3985729651634
// MI455X (gfx1250) — compile-verified
//
#include <hip/hip_runtime.h>

typedef __attribute__((ext_vector_type(2))) float v2f;
typedef __attribute__((ext_vector_type(8))) float v8f;

namespace {
constexpr int B_ = 4096;
constexpr int T_ = 96;
constexpr int N_ = 32;
constexpr int E_ = 4;
constexpr int TN = T_ * N_;          // 3072
constexpr int FROWS = 98;            // 49 cos rows + 49 sin rows (rows >=98 are zero)
constexpr int AROWS = 99;            // 98 real rows + one shared all-zero row
constexpr int ULEN = AROWS * 96;     // 9504 floats: Amat (Phase B) then Wexp (Phase F)
constexpr float W0 = 6.2831853071795864769f / 96.0f;
}

struct KParams {
  const float* x;        // (B,T,N,1)
  const float* noise;    // (B,E)
  const float* w_start;  // (N)
  const float* b_start;  // (1)
  const float* wg;       // (E,T)
  const float* bg;       // (E)
  const float* wn;       // (E,T)
  const float* bn;       // (E)
  const float* ew[16];   // e{0..3}_{w1,b1,w2,b2}
};

// gelu(tanh approx): 0.5*v*(1+tanh(u)) == v * sigmoid(2u); native exp only.
__device__ __forceinline__ float gelu_tanh(float v) {
  const float c = 0.7978845608028654f;  // sqrt(2/pi)
  float u = c * (v + 0.044715f * v * v * v);
  return v / (1.0f + __expf(-2.0f * u));
}

__device__ __forceinline__ int clamp95(int i) { return i < 0 ? 0 : (i > 95 ? 95 : i); }

// One patch-expert MLP: y = gelu(xr @ w1 + b1) @ w2 + b2, accumulated *g into OutS.
// Each thread handles TWO adjacent n columns so every lane-uniform LDS weight
// broadcast feeds two FMA chains (halves ds traffic per flop).
template <int P>
__device__ void run_expert(const float* __restrict__ w1, const float* __restrict__ b1,
                           const float* __restrict__ w2, const float* __restrict__ b2,
                           const float* __restrict__ Xs, float* __restrict__ OutS, float g,
                           int tid) {
  if (g == 0.0f) return;  // block-uniform: only top-K experts have nonzero gates
  constexpr int C = 96 / P;
  constexpr int ROWS = C * 16;  // n handled in pairs
  for (int task = tid; task < ROWS; task += 256) {
    int c = task >> 4;
    int n0 = (task & 15) * 2;
    float xv0[P], xv1[P];
#pragma unroll
    for (int j = 0; j < P; ++j) {
      int o = (c * P + j) * 32 + n0;
      xv0[j] = Xs[o];
      xv1[j] = Xs[o + 1];
    }
    float h0[32], h1[32];
#pragma unroll
    for (int d = 0; d < 32; ++d) {
      float a0 = b1[d], a1 = a0;
#pragma unroll
      for (int j = 0; j < P; ++j) {
        float w = w1[j * 32 + d];
        a0 = fmaf(xv0[j], w, a0);
        a1 = fmaf(xv1[j], w, a1);
      }
      h0[d] = gelu_tanh(a0);
      h1[d] = gelu_tanh(a1);
    }
#pragma unroll
    for (int r = 0; r < P; ++r) {
      float a0 = b2[r], a1 = a0;
#pragma unroll
      for (int d = 0; d < 32; ++d) {
        float w = w2[d * P + r];
        a0 = fmaf(h0[d], w, a0);
        a1 = fmaf(h1[d], w, a1);
      }
      int o = (c * P + r) * 32 + n0;
      OutS[o] += g * a0;
      OutS[o + 1] += g * a1;
    }
  }
}

__global__ __launch_bounds__(256) void moe_forward_kernel(KParams P, float* __restrict__ out,
                                                          float* __restrict__ ws) {
  __shared__ float Xs[TN];            // x2 tile, t-major / n inner            (12288 B)
  __shared__ float Fs[FROWS * 32];    // DFT result; later Out accumulator     (12544 B)
  __shared__ float Ubuf[ULEN];        // Amat (Phase B) / Wexp (Phase F)       (38016 B)
  __shared__ float Trig[288];         // cos | sin | zeros, step W0            ( 1152 B)
  __shared__ float xlin[T_];
  __shared__ float cleanS[E_], stdS[E_], noisyS[E_], gatesS[E_];
  __shared__ int fidxS[N_ * 3];
  __shared__ float fsclS[N_ * 3];

  const int b = blockIdx.x;
  const int tid = threadIdx.x;
  const int lane = tid & 31;
  const int wave = tid >> 5;

  // ---- Phase A: stage x tile + trig table into LDS ----
  const float* xg = P.x + (size_t)b * TN;
  for (int i = tid; i < TN; i += 256) Xs[i] = xg[i];
  if (tid < 96) {
    float ang = W0 * (float)tid;
    Trig[tid] = __cosf(ang);
    Trig[96 + tid] = __sinf(ang);
    Trig[192 + tid] = 0.0f;
  }
  __syncthreads();

  // ---- Phase A2: materialize DFT coefficient matrix Amat[99][96] in LDS ----
  // rows 0..48: cos(W0*r*t), rows 49..97: sin(W0*(r-49)*t), row 98: zeros.
  {
    float* Amat = Ubuf;
    for (int i = tid; i < AROWS * 96; i += 256) {
      int r = i / 96;
      int t = i - r * 96;
      float v;
      if (r < 49) v = Trig[(r * t) % 96];
      else if (r < 98) v = Trig[96 + ((r - 49) * t) % 96];
      else v = 0.0f;
      Amat[i] = v;
    }
  }
  __syncthreads();

  // ---- Phase B: forward real DFT as f32 WMMA GEMM: F = A[112x96] * Xs[96x32] ----
  // 14 output tiles (7 row-tiles x 2 col-tiles) over 8 waves; 24 fully-unrolled
  // K-chunks. A fragment = one ds_load_b64 at an immediate offset (no index math).
  const int khalf = lane >> 4;  // A/B fragment layout: v0 holds K={0,2}, v1 holds K={1,3}
  {
    const float* Amat = Ubuf;
    for (int tile = wave; tile < 14; tile += 8) {
      const int rt = tile >> 1;
      const int ct = tile & 1;
      const int r = rt * 16 + (lane & 15);
      const int rmin = r < 98 ? r : 98;              // rows >=98 read the zero row
      const int col = ct * 16 + (lane & 15);
      const int abase = rmin * 96 + khalf * 2;
      const int bbase = (khalf * 2) * 32 + col;
      v8f acc = {};
#pragma unroll
      for (int kc = 0; kc < 24; ++kc) {
        v2f a = *(const v2f*)&Amat[abase + kc * 4];
        v2f bb;
        bb.x = Xs[bbase + kc * 128];
        bb.y = Xs[bbase + kc * 128 + 32];
        acc = __builtin_amdgcn_wmma_f32_16x16x4_f32(false, a, false, bb, (short)0, acc, false,
                                                    false);
      }
      const int rbase = rt * 16 + ((lane >= 16) ? 8 : 0);
#pragma unroll
      for (int v = 0; v < 8; ++v)
        if (rbase + v < FROWS) Fs[(rbase + v) * 32 + col] = acc[v];
    }
  }
  __syncthreads();

  // ---- Phase A3: Amat is dead; stage packed expert weights into the union ----
  {
    float* Wexp = Ubuf;
    constexpr int segsz[16] = {256, 32, 256, 8, 192, 32, 192, 6,
                               128, 32, 128, 4, 384, 32, 384, 12};
    constexpr int segoff[16] = {0,   256,  288,  544,  552,  744,  776,  968,
                                974, 1102, 1134, 1262, 1266, 1650, 1682, 2066};
#pragma unroll
    for (int s = 0; s < 16; ++s)
      for (int i = tid; i < segsz[s]; i += 256) Wexp[segoff[s] + i] = P.ew[s][i];
  }

  // ---- Phase C: per-channel top-3 spectral bins (f=0 excluded) ----
  if (tid < N_) {
    int n = tid;
    float p0 = -1.f, p1 = -1.f, p2 = -1.f;
    int i0 = 1, i1 = 1, i2 = 1;
    for (int f = 1; f <= 48; ++f) {
      float cr = Fs[f * 32 + n];
      float si = Fs[(49 + f) * 32 + n];
      float p = cr * cr + si * si;
      if (p > p0) {
        p2 = p1; i2 = i1; p1 = p0; i1 = i0; p0 = p; i0 = f;
      } else if (p > p1) {
        p2 = p1; i2 = i1; p1 = p; i1 = f;
      } else if (p > p2) {
        p2 = p; i2 = f;
      }
    }
    fidxS[n * 3 + 0] = i0;
    fidxS[n * 3 + 1] = i1;
    fidxS[n * 3 + 2] = i2;
    fsclS[n * 3 + 0] = (i0 == 48 ? 1.f : 2.f) / 96.f;
    fsclS[n * 3 + 1] = (i1 == 48 ? 1.f : 2.f) / 96.f;
    fsclS[n * 3 + 2] = (i2 == 48 ? 1.f : 2.f) / 96.f;
  }
  __syncthreads();

  // ---- Phase D: season + trend + x, reduce over n -> x_lin[t] ----
  const float bstart0 = P.b_start[0];
  for (int base = 0; base < TN; base += 256) {
    int idx = base + tid;
    int t = idx >> 5, n = idx & 31;
    float xv = Xs[idx];
    float season = 0.0f;
#pragma unroll
    for (int j = 0; j < 3; ++j) {
      int f = fidxS[n * 3 + j];
      float sc = fsclS[n * 3 + j];
      int m = (f * t) % 96;
      season += sc * (Fs[f * 32 + n] * Trig[m] + Fs[(49 + f) * 32 + n] * Trig[96 + m]);
    }
    float s4 = 0.f, s8 = 0.f, s12 = 0.f;
    for (int i = t - 1; i <= t + 2; ++i) s4 += Xs[clamp95(i) * 32 + n];
    for (int i = t - 3; i <= t + 4; ++i) s8 += Xs[clamp95(i) * 32 + n];
    for (int i = t - 5; i <= t + 6; ++i) s12 += Xs[clamp95(i) * 32 + n];
    float trend = (s4 * 0.25f + s8 * 0.125f + s12 * (1.0f / 12.0f)) * (1.0f / 3.0f);
    float v = xv + season + trend;
    float contrib = v * P.w_start[n];
    for (int off = 16; off > 0; off >>= 1) contrib += __shfl_down(contrib, off, 32);
    if (lane == 0) xlin[t] = contrib + bstart0;  // one wave == one t
  }
  __syncthreads();

  // ---- Phase E: noisy top-k gating ----
  if (tid < E_) {
    int e = tid;
    float c = P.bg[e], s = P.bn[e];
    const float* wgr = P.wg + e * T_;
    const float* wnr = P.wn + e * T_;
    for (int t = 0; t < T_; ++t) {
      float xl = xlin[t];
      c = fmaf(xl, wgr[t], c);
      s = fmaf(xl, wnr[t], s);
    }
    float sp = fmaxf(s, 0.0f) + log1pf(__expf(-fabsf(s)));  // softplus
    float sd = fminf(fmaxf(sp + 0.01f, 0.001f), 1000.0f);
    cleanS[e] = c;
    stdS[e] = sd;
    noisyS[e] = c + P.noise[(size_t)b * E_ + e] * sd;
  }
  __syncthreads();

  if (tid == 0) {
    float v[4];
    int id[4];
    for (int e = 0; e < 4; ++e) { v[e] = noisyS[e]; id[e] = e; }
    for (int i = 0; i < 3; ++i)
      for (int j = i + 1; j < 4; ++j)
        if (v[j] > v[i]) {
          float tv = v[i]; v[i] = v[j]; v[j] = tv;
          int ti = id[i]; id[i] = id[j]; id[j] = ti;
        }
    float m = fmaxf(v[0], v[1]);
    float e0 = __expf(v[0] - m), e1 = __expf(v[1] - m);
    float inv = 1.0f / (e0 + e1);
    float g[4] = {0.f, 0.f, 0.f, 0.f};
    g[id[0]] = e0 * inv;
    g[id[1]] = e1 * inv;
    const float thr_in = v[2], thr_out = v[1];
    for (int e = 0; e < 4; ++e) {
      float sd = fmaxf(stdS[e], 1e-6f);
      float zi = fminf(fmaxf((cleanS[e] - thr_in) / sd, -50.f), 50.f);
      float zo = fminf(fmaxf((cleanS[e] - thr_out) / sd, -50.f), 50.f);
      float pi = fminf(fmaxf(0.5f * (1.0f + erff(zi * 0.70710678118654752f)), 0.f), 1.f);
      float po = fminf(fmaxf(0.5f * (1.0f + erff(zo * 0.70710678118654752f)), 0.f), 1.f);
      float p = (noisyS[e] > thr_in) ? pi : po;
      atomicAdd(&ws[e], p);        // load
      atomicAdd(&ws[4 + e], g[e]); // importance
      gatesS[e] = g[e];
    }
  }

  // ---- Phase F: gated patch experts; reuse Fs region as out accumulator ----
  float* OutS = Fs;
  float* Wexp = Ubuf;
  for (int i = tid; i < TN; i += 256) OutS[i] = 0.0f;
  __syncthreads();
  run_expert<8>(&Wexp[0], &Wexp[256], &Wexp[288], &Wexp[544], Xs, OutS, gatesS[0], tid);
  __syncthreads();
  run_expert<6>(&Wexp[552], &Wexp[744], &Wexp[776], &Wexp[968], Xs, OutS, gatesS[1], tid);
  __syncthreads();
  run_expert<4>(&Wexp[974], &Wexp[1102], &Wexp[1134], &Wexp[1262], Xs, OutS, gatesS[2], tid);
  __syncthreads();
  run_expert<12>(&Wexp[1266], &Wexp[1650], &Wexp[1682], &Wexp[2066], Xs, OutS, gatesS[3], tid);
  __syncthreads();

  // ---- Phase G: coalesced writeback ----
  float* og = out + (size_t)b * TN;
  for (int i = tid; i < TN; i += 256) og[i] = OutS[i];
}

__device__ __forceinline__ float cv_squared4(const float* v) {
  float m = 0.25f * (v[0] + v[1] + v[2] + v[3]);
  float var = 0.f;
  for (int i = 0; i < 4; ++i) {
    float d = v[i] - m;
    var += d * d;
  }
  var *= (1.0f / 3.0f);  // ddof=1
  return var / (m * m + 1e-10f);
}

__global__ void moe_zero_ws(float* ws) {
  int i = threadIdx.x;
  if (i < 8) ws[i] = 0.0f;
}

__global__ void moe_finalize_kernel(const float* __restrict__ ws, float* __restrict__ loss_out) {
  if (threadIdx.x == 0 && blockIdx.x == 0) {
    float ld[4] = {ws[0], ws[1], ws[2], ws[3]};
    float imp[4] = {ws[4], ws[5], ws[6], ws[7]};
    *loss_out = (cv_squared4(imp) + cv_squared4(ld)) * 1e-4f;
  }
}

extern "C" void kernel_launch(void* const* d_in, const int* in_sizes, int n_in, void* d_out,
                              int out_size, void* d_ws, size_t ws_size, hipStream_t stream) {
  (void)in_sizes; (void)n_in; (void)out_size; (void)ws_size;
  KParams P;
  P.x = (const float*)d_in[0];
  P.noise = (const float*)d_in[1];
  P.w_start = (const float*)d_in[2];
  P.b_start = (const float*)d_in[3];
  P.wg = (const float*)d_in[4];
  P.bg = (const float*)d_in[5];
  P.wn = (const float*)d_in[6];
  P.bn = (const float*)d_in[7];
  for (int i = 0; i < 16; ++i) P.ew[i] = (const float*)d_in[8 + i];

  float* out = (float*)d_out;
  float* ws = (float*)d_ws;

  moe_zero_ws<<<1, 32, 0, stream>>>(ws);
  moe_forward_kernel<<<B_, 256, 0, stream>>>(P, out, ws);
  moe_finalize_kernel<<<1, 1, 0, stream>>>(ws, out + (size_t)B_ * TN);
}